// ENNWithBICEP_22445499089478
// MI455X (gfx1250) — compile-verified
//
#include <hip/hip_runtime.h>
#include <hip/hip_bf16.h>

// ---------------------------------------------------------------------------
// ENN+BICEP forward for MI455X (gfx1250, wave32).
//  - All dense layers: fp32 WMMA (v_wmma_f32_16x16x4_f32); wave computes a
//    32x16 strip; K in groups of 16 (batched loads); single-use weight
//    streams use non-temporal loads to spare the 192MB L2.
//  - LSTM: input half of the gates hoisted into one WMMA GEMM over all
//    32000 timesteps; the sequential scan only does the recurrent half.
//  - Attention: fused flash-style per (b, q-tile) WG, K/V staged to LDS via
//    gfx1250 async global->LDS loads (ASYNCcnt) when available, fp32 WMMA
//    for QK^T and PV, head-mean accumulated in LDS.
//  - BICEP SDE+conv: linear recurrence collapsed to a 3-coefficient dot
//    against the noise stream (reads 82MB noise once, no paths buffer).
// ---------------------------------------------------------------------------

typedef __attribute__((ext_vector_type(2))) float v2f;
typedef __attribute__((ext_vector_type(8))) float v8f;

__device__ __forceinline__ v8f wmma4(v2f a, v2f b, v8f c) {
  // D = A(16x4) * B(4x16) + C(16x16), fp32
  return __builtin_amdgcn_wmma_f32_16x16x4_f32(false, a, false, b, (short)0, c,
                                               false, false);
}

#if __has_builtin(__builtin_amdgcn_global_load_async_to_lds_b32) && \
    __has_builtin(__builtin_amdgcn_s_wait_asynccnt)
#define USE_ASYNC_LDS 1
typedef __attribute__((address_space(1))) int as1_int;
typedef __attribute__((address_space(3))) int as3_int;
__device__ __forceinline__ void async_g2l_b32(const float* g, float* l) {
  __builtin_amdgcn_global_load_async_to_lds_b32((as1_int*)g, (as3_int*)l, 0,
                                                0);
}
#else
#define USE_ASYNC_LDS 0
#endif

// ---------------------------------------------------------------------------
// Generic  C[M x N] = A[M x K] * W[N x K]^T (+bias) (+accumulate into C)
// Requires M % 32 == 0, N % 16 == 0, K % 4 == 0.
// wnt != 0: stream W with non-temporal loads (single-use weight matrices).
// ---------------------------------------------------------------------------
__global__ void gemm_wT_kernel(const float* __restrict__ A, int lda,
                               const float* __restrict__ W, int ldw,
                               const float* __restrict__ bias,
                               float* __restrict__ C, int ldc,
                               int M, int N, int K, int acc, int wnt) {
  const int lane = threadIdx.x & 31;
  const int wid = blockIdx.x * (blockDim.x >> 5) + (threadIdx.x >> 5);
  const int Nt = N >> 4;
  const int Mp = M >> 5;
  if (wid >= Mp * Nt) return;
  const int nt = wid % Nt;
  const int mp = wid / Nt;
  const int n0 = nt << 4;
  const int m0 = mp << 5;
  const int l15 = lane & 15;
  const int hi = lane >> 4;

  const float* Arow0 = A + (size_t)(m0 + l15) * lda + 2 * hi;
  const float* Arow1 = Arow0 + (size_t)16 * lda;
  const float* Wrow = W + (size_t)(n0 + l15) * ldw + 2 * hi;

  v8f c0 = {};
  v8f c1 = {};
  if (acc) {
#pragma unroll
    for (int r = 0; r < 8; ++r) {
      int row = m0 + r + 8 * hi;
      c0[r] = C[(size_t)row * ldc + n0 + l15];
      c1[r] = C[(size_t)(row + 16) * ldc + n0 + l15];
    }
  }

  const int Kmain = K & ~15;
  if (wnt) {
    for (int kc = 0; kc < Kmain; kc += 16) {
      v2f bb[4], a0v[4], a1v[4];
#pragma unroll
      for (int u = 0; u < 4; ++u) {
        bb[u] = __builtin_nontemporal_load((const v2f*)(Wrow + kc + 4 * u));
        a0v[u] = *(const v2f*)(Arow0 + kc + 4 * u);
        a1v[u] = *(const v2f*)(Arow1 + kc + 4 * u);
      }
#pragma unroll
      for (int u = 0; u < 4; ++u) {
        c0 = wmma4(a0v[u], bb[u], c0);
        c1 = wmma4(a1v[u], bb[u], c1);
      }
    }
  } else {
    for (int kc = 0; kc < Kmain; kc += 16) {
      v2f bb[4], a0v[4], a1v[4];
#pragma unroll
      for (int u = 0; u < 4; ++u) {
        bb[u] = *(const v2f*)(Wrow + kc + 4 * u);
        a0v[u] = *(const v2f*)(Arow0 + kc + 4 * u);
        a1v[u] = *(const v2f*)(Arow1 + kc + 4 * u);
      }
#pragma unroll
      for (int u = 0; u < 4; ++u) {
        c0 = wmma4(a0v[u], bb[u], c0);
        c1 = wmma4(a1v[u], bb[u], c1);
      }
    }
  }
  for (int kc = Kmain; kc < K; kc += 4) {
    v2f b = *(const v2f*)(Wrow + kc);
    v2f a0 = *(const v2f*)(Arow0 + kc);
    v2f a1 = *(const v2f*)(Arow1 + kc);
    c0 = wmma4(a0, b, c0);
    c1 = wmma4(a1, b, c1);
  }

  const float bv = bias ? bias[n0 + l15] : 0.0f;
#pragma unroll
  for (int r = 0; r < 8; ++r) {
    int row = m0 + r + 8 * hi;
    C[(size_t)row * ldc + n0 + l15] = c0[r] + bv;
    C[(size_t)(row + 16) * ldc + n0 + l15] = c1[r] + bv;
  }
}

// bsum[i] = b_ih[i] + b_hh[i]
__global__ void biasadd_kernel(const float* __restrict__ a,
                               const float* __restrict__ b,
                               float* __restrict__ o) {
  int i = threadIdx.x;
  o[i] = a[i] + b[i];
}

// ---------------------------------------------------------------------------
// LSTM recurrent half: one block per batch row, 256 threads (one gate each).
// xg = precomputed ns@w_ih^T + b_ih + b_hh, layout [32000, 256].
// w_hh rows in registers; h broadcast via LDS; c in registers.
// ---------------------------------------------------------------------------
__global__ void lstm_kernel(const float* __restrict__ xg,
                            const float* __restrict__ w_hh,
                            float* __restrict__ lstm_out) {
  __shared__ __align__(16) float hsh[64];
  __shared__ float gates[256];
  const int g = threadIdx.x;
  const int b = blockIdx.x;
  float whh[64];
#pragma unroll
  for (int s = 0; s < 64; ++s) whh[s] = w_hh[g * 64 + s];
  float cst = 0.0f;
  if (g < 64) hsh[g] = 0.0f;
  const float* xgb = xg + (size_t)b * 1000 * 256;
  float xcur = xgb[g];
  float xnxt = xgb[256 + g];
  __syncthreads();
  for (int n = 0; n < 1000; ++n) {
    if (n + 8 < 1000) __builtin_prefetch(xgb + (size_t)(n + 8) * 256 + g, 0, 0);
    float acc[4];
    acc[0] = xcur;
    acc[1] = 0.0f;
    acc[2] = 0.0f;
    acc[3] = 0.0f;
#pragma unroll
    for (int s4 = 0; s4 < 16; ++s4) {
      float4 hv = *(const float4*)&hsh[s4 * 4];
      acc[s4 & 3] += hv.x * whh[4 * s4 + 0] + hv.y * whh[4 * s4 + 1] +
                     hv.z * whh[4 * s4 + 2] + hv.w * whh[4 * s4 + 3];
    }
    gates[g] = (acc[0] + acc[1]) + (acc[2] + acc[3]);
    float xn2 = (n + 2 < 1000) ? xgb[(size_t)(n + 2) * 256 + g] : 0.0f;
    __syncthreads();
    if (g < 64) {
      float ig = gates[g], fg = gates[64 + g], gg = gates[128 + g],
            og = gates[192 + g];
      float si = 1.0f / (1.0f + __expf(-ig));
      float sf = 1.0f / (1.0f + __expf(-fg));
      float so = 1.0f / (1.0f + __expf(-og));
      cst = sf * cst + si * tanhf(gg);
      float h = so * tanhf(cst);
      hsh[g] = h;
      lstm_out[(size_t)(b * 1000 + n) * 64 + g] = h;
    }
    xcur = xnxt;
    xnxt = xn2;
    __syncthreads();
  }
}

// ---------------------------------------------------------------------------
// Fused attention: one block per (batch, 16-query tile). 8 waves.
// qkv layout: [b*1000+n][192] (q: cols 0..63, k: 64..127, v: 128..191)
// ---------------------------------------------------------------------------
#define KS 18    // kbuf row stride (keys x dims, padded)
#define SS 1012  // sbuf row stride
#define VS 1012  // vbufT row stride

__global__ void attn_kernel(const float* __restrict__ qkv,
                            float* __restrict__ ctx,
                            float* __restrict__ attn_out) {
  extern __shared__ float sm[];
  float* kbuf = sm;                 // 1008*18
  float* vbufT = kbuf + 1008 * KS;  // 16*1012
  float* sbuf = vbufT + 16 * VS;    // 16*1012
  float* meanb = sbuf + 16 * SS;    // 16*1000
  float* redb = meanb + 16 * 1000;  // 8*256
  const int t = threadIdx.x;
  const int lane = t & 31, w = t >> 5;
  const int l15 = lane & 15, hi = lane >> 4;
  const int b = blockIdx.x / 63;
  const int q0 = (blockIdx.x % 63) * 16;

  for (int i = t; i < 16 * 1000; i += 256) meanb[i] = 0.0f;
  __syncthreads();

  for (int h = 0; h < 4; ++h) {
    // stage K (row-major) and V (transposed) for this head
    for (int i = t; i < 1008 * 16; i += 256) {
      int key = i >> 4, d = i & 15;
      if (key < 1000) {
        const float* src =
            qkv + (size_t)(b * 1000 + key) * 192 + h * 16 + d;
#if USE_ASYNC_LDS
        async_g2l_b32(src + 64, &kbuf[key * KS + d]);
        async_g2l_b32(src + 128, &vbufT[d * VS + key]);
#else
        kbuf[key * KS + d] = src[64];
        vbufT[d * VS + key] = src[128];
#endif
      } else {
        kbuf[key * KS + d] = 0.0f;
        vbufT[d * VS + key] = 0.0f;
      }
    }
#if USE_ASYNC_LDS
    __builtin_amdgcn_s_wait_asynccnt(0);
#endif
    __syncthreads();

    // Q fragments for this wave's query tile (zeros past row 999)
    v2f aq[4];
    {
      int qrow = q0 + l15;
#pragma unroll
      for (int kc4 = 0; kc4 < 4; ++kc4) {
        if (qrow < 1000) {
          aq[kc4] = *(const v2f*)(qkv + (size_t)(b * 1000 + qrow) * 192 +
                                  h * 16 + kc4 * 4 + 2 * hi);
        } else {
          aq[kc4][0] = 0.0f;
          aq[kc4][1] = 0.0f;
        }
      }
    }
    // scores: S = (Q K^T) * 1/sqrt(16)
    for (int nt = w; nt < 63; nt += 8) {
      int n0 = nt << 4;
      v2f bf[4];
#pragma unroll
      for (int kc4 = 0; kc4 < 4; ++kc4)
        bf[kc4] = *(const v2f*)(kbuf + (n0 + l15) * KS + kc4 * 4 + 2 * hi);
      v8f c = {};
#pragma unroll
      for (int kc4 = 0; kc4 < 4; ++kc4) c = wmma4(aq[kc4], bf[kc4], c);
#pragma unroll
      for (int r = 0; r < 8; ++r)
        sbuf[(r + 8 * hi) * SS + n0 + l15] = c[r] * 0.25f;
    }
    __syncthreads();

    // softmax: wave w owns rows 2w, 2w+1 (16 queries, 1000 keys)
    for (int rr = 0; rr < 2; ++rr) {
      int row = 2 * w + rr;
      float mx = -3.0e38f;
      for (int i = lane; i < 1000; i += 32)
        mx = fmaxf(mx, sbuf[row * SS + i]);
      for (int m = 16; m > 0; m >>= 1) mx = fmaxf(mx, __shfl_xor(mx, m, 32));
      float ssum = 0.0f;
      for (int i = lane; i < 1000; i += 32) {
        float e = __expf(sbuf[row * SS + i] - mx);
        sbuf[row * SS + i] = e;
        ssum += e;
      }
      for (int m = 16; m > 0; m >>= 1) ssum += __shfl_xor(ssum, m, 32);
      float inv = 1.0f / ssum;
      for (int i = lane; i < 1000; i += 32) {
        float p = sbuf[row * SS + i] * inv;
        sbuf[row * SS + i] = p;
        meanb[row * 1000 + i] += 0.25f * p;
      }
    }
    __syncthreads();

    // ctx = P @ V : each wave covers a K-range of keys, then LDS reduce
    v8f c = {};
    int kcBeg = w * 128;
    int kcEnd = (kcBeg + 128 < 1000) ? (kcBeg + 128) : 1000;
    for (int kc = kcBeg; kc < kcEnd; kc += 8) {
      v2f af0 = *(const v2f*)(sbuf + l15 * SS + kc + 2 * hi);
      v2f bf0 = *(const v2f*)(vbufT + l15 * VS + kc + 2 * hi);
      v2f af1 = *(const v2f*)(sbuf + l15 * SS + kc + 4 + 2 * hi);
      v2f bf1 = *(const v2f*)(vbufT + l15 * VS + kc + 4 + 2 * hi);
      c = wmma4(af0, bf0, c);
      c = wmma4(af1, bf1, c);
    }
#pragma unroll
    for (int r = 0; r < 8; ++r) redb[w * 256 + r * 32 + lane] = c[r];
    __syncthreads();
    {
      int r = t >> 5;
      int ln = t & 31;
      float sum = 0.0f;
#pragma unroll
      for (int ww = 0; ww < 8; ++ww) sum += redb[ww * 256 + r * 32 + ln];
      int row = r + 8 * (ln >> 4);
      int col = ln & 15;
      int qrow = q0 + row;
      if (qrow < 1000)
        ctx[(size_t)(b * 1000 + qrow) * 64 + h * 16 + col] = sum;
    }
    __syncthreads();
  }
  // write head-averaged attention map
  for (int i = t; i < 16 * 1000; i += 256) {
    int r = i / 1000, col = i - r * 1000;
    int qrow = q0 + r;
    if (qrow < 1000)
      attn_out[(size_t)(b * 1000 + qrow) * 1000 + col] = meanb[i];
  }
}

// ---------------------------------------------------------------------------
// enn_pre[b,s] = relu(mean_n attended[b,n,s])
// ---------------------------------------------------------------------------
__global__ void ennpre_kernel(const float* __restrict__ att,
                              float* __restrict__ ennpre) {
  __shared__ float red[256];
  int b = blockIdx.x, t = threadIdx.x;
  int s = t & 63, part = t >> 6;
  float sum = 0.0f;
  for (int n = part; n < 1000; n += 4)
    sum += att[(size_t)(b * 1000 + n) * 64 + s];
  red[t] = sum;
  __syncthreads();
  if (t < 64) {
    float v = red[t] + red[64 + t] + red[128 + t] + red[192 + t];
    v *= (1.0f / 1000.0f);
    ennpre[b * 64 + t] = fmaxf(v, 0.0f);
  }
}

// ---------------------------------------------------------------------------
// BICEP closed form: S_k[t] = sum_{c=t+1}^{640} conv_w[c,k] * a^(c-1-t)
// ---------------------------------------------------------------------------
__global__ void sdecoef_kernel(const float* __restrict__ conv_w,
                               const float* __restrict__ decay,
                               float* __restrict__ S) {
  int k = threadIdx.x;
  if (k >= 3) return;
  const float dt = 1.0f / 640.0f;
  const float a = 1.0f - decay[0] * dt;
  float s = 0.0f;
  for (int tt = 639; tt >= 0; --tt) {
    s = conv_w[(tt + 1) * 3 + k] + a * s;
    S[k * 640 + tt] = s;
  }
}

// fb[b,j] = sigmoid(x[b,:] . W_inproj[j,:] + b_inproj[j]) for j < 10
__global__ void fb_kernel(const float* __restrict__ x,
                          const float* __restrict__ W_inproj,
                          const float* __restrict__ b_inproj,
                          float* __restrict__ fb) {
  int lane = threadIdx.x & 31;
  int wv = blockIdx.x * 8 + (threadIdx.x >> 5);
  if (wv >= 320) return;
  int b = wv / 10, j = wv % 10;
  const float* xr = x + (size_t)b * 1024;
  const float* wr = W_inproj + (size_t)j * 1024;
  float sum = 0.0f;
  for (int i = lane; i < 1024; i += 32) sum += xr[i] * wr[i];
  for (int m = 16; m > 0; m >>= 1) sum += __shfl_xor(sum, m, 32);
  if (lane == 0) fb[b * 10 + j] = 1.0f / (1.0f + __expf(-(sum + b_inproj[j])));
}

// G_k[b,m] = sq * f[b,m/100] * sum_t noise[b,m,t] * S_k[t]  (wave per row)
__global__ void bicep_reduce_kernel(const float* __restrict__ noise,
                                    const float* __restrict__ S,
                                    const float* __restrict__ fb,
                                    float* __restrict__ G) {
  __shared__ __align__(16) float Sl[1920];
  int t = threadIdx.x;
  for (int i = t; i < 1920; i += 256) Sl[i] = S[i];
  __syncthreads();
  int lane = t & 31;
  int row = blockIdx.x * 8 + (t >> 5);
  if (row >= 32000) return;
  const float* nr = noise + (size_t)row * 640;
  float a0 = 0.0f, a1 = 0.0f, a2 = 0.0f;
#pragma unroll
  for (int c = 0; c < 5; ++c) {
    int t0 = c * 128 + lane * 4;
    float4 nv = *(const float4*)(nr + t0);
    float4 s0 = *(const float4*)(Sl + t0);
    float4 s1 = *(const float4*)(Sl + 640 + t0);
    float4 s2 = *(const float4*)(Sl + 1280 + t0);
    a0 += nv.x * s0.x + nv.y * s0.y + nv.z * s0.z + nv.w * s0.w;
    a1 += nv.x * s1.x + nv.y * s1.y + nv.z * s1.z + nv.w * s1.w;
    a2 += nv.x * s2.x + nv.y * s2.y + nv.z * s2.z + nv.w * s2.w;
  }
  for (int m = 16; m > 0; m >>= 1) {
    a0 += __shfl_xor(a0, m, 32);
    a1 += __shfl_xor(a1, m, 32);
    a2 += __shfl_xor(a2, m, 32);
  }
  if (lane == 0) {
    int b = row / 1000, n = row - b * 1000;
    const float sq = 0.03952847075210474f;  // sqrt(1/640)
    float scale = sq * fb[b * 10 + n / 100];
    G[row] = scale * a0;
    G[32000 + row] = scale * a1;
    G[64000 + row] = scale * a2;
  }
}

// agg[b,n] = G0[b,n-1] + G1[b,n] + G2[b,n+1] + conv_b  (zero-padded edges)
__global__ void agg_kernel(const float* __restrict__ G,
                           const float* __restrict__ conv_b,
                           float* __restrict__ agg) {
  int i = blockIdx.x * 256 + threadIdx.x;
  if (i >= 32000) return;
  int n = i % 1000;
  float v = G[32000 + i] + conv_b[0];
  if (n > 0) v += G[i - 1];
  if (n < 999) v += G[64000 + i + 1];
  agg[i] = v;
}

// ---------------------------------------------------------------------------
// Host-side orchestration
// ---------------------------------------------------------------------------
static inline void launch_gemm(const float* A, int lda, const float* W, int ldw,
                               const float* bias, float* C, int ldc, int M,
                               int N, int K, int acc, int wnt, hipStream_t s) {
  int waves = (M / 32) * (N / 16);
  int blocks = (waves + 7) / 8;
  gemm_wT_kernel<<<blocks, 256, 0, s>>>(A, lda, W, ldw, bias, C, ldc, M, N, K,
                                        acc, wnt);
}

extern "C" void kernel_launch(void* const* d_in, const int* in_sizes, int n_in,
                              void* d_out, int out_size, void* d_ws,
                              size_t ws_size, hipStream_t stream) {
  (void)in_sizes; (void)n_in; (void)out_size; (void)ws_size;
  const float* x        = (const float*)d_in[0];
  const float* noise    = (const float*)d_in[1];
  const float* W_emb    = (const float*)d_in[2];
  const float* b_emb    = (const float*)d_in[3];
  const float* w_ih     = (const float*)d_in[4];
  const float* w_hh     = (const float*)d_in[5];
  const float* b_ih     = (const float*)d_in[6];
  const float* b_hh     = (const float*)d_in[7];
  const float* w_in     = (const float*)d_in[8];
  const float* b_in     = (const float*)d_in[9];
  const float* w_out    = (const float*)d_in[10];
  const float* b_out    = (const float*)d_in[11];
  const float* W_fix    = (const float*)d_in[12];
  const float* W_inproj = (const float*)d_in[13];
  const float* b_inproj = (const float*)d_in[14];
  const float* conv_w   = (const float*)d_in[15];
  const float* conv_b   = (const float*)d_in[16];
  const float* W_bout   = (const float*)d_in[17];
  const float* b_bout   = (const float*)d_in[18];
  const float* decay    = (const float*)d_in[19];
  const float* W_fuse   = (const float*)d_in[20];
  const float* b_fuse   = (const float*)d_in[21];
  const float* W_proj   = (const float*)d_in[22];
  const float* b_proj   = (const float*)d_in[23];

  float* out      = (float*)d_out;            // [32,1024]
  float* enn      = out + 32768;              // [32,1024]
  float* bicep    = out + 65536;              // [32,1024]
  float* attn_out = out + 98304;              // [32,1000,1000]

  float* ws = (float*)d_ws;
  float* ns     = ws + 0;         // [32,64000]  (later reused as ctx [32000,64])
  float* lstm   = ws + 2048000;   // [32000,64]  (later reused as attended)
  float* qkv    = ws + 4096000;   // [32000,192]
  float* xg     = ws + 4096000;   // [32000,256] aliases qkv (dead before qkv)
  float* Sc     = ws + 12288000;  // [3,640]
  float* G      = ws + 12290048;  // [3,32000]
  float* agg    = ws + 12388352;  // [32,1000]
  float* fbuf   = ws + 12421120;  // [32,10]
  float* ennpre = ws + 12421632;  // [32,64]
  float* bsum   = ws + 12423680;  // [256]
  float* fused  = ws + 12424192;  // [32,1024]
  float* ctx = ns;
  float* attended = lstm;

  // ---- BICEP branch (independent of ENN chain) ----
  sdecoef_kernel<<<1, 32, 0, stream>>>(conv_w, decay, Sc);
  fb_kernel<<<40, 256, 0, stream>>>(x, W_inproj, b_inproj, fbuf);
  bicep_reduce_kernel<<<4000, 256, 0, stream>>>(noise, Sc, fbuf, G);
  agg_kernel<<<125, 256, 0, stream>>>(G, conv_b, agg);
  launch_gemm(agg, 1000, W_bout, 1000, b_bout, bicep, 1024, 32, 1024, 1000, 0,
              1, stream);

  // ---- ENN branch ----
  launch_gemm(x, 1024, W_emb, 1024, b_emb, ns, 64000, 32, 64000, 1024, 0, 1,
              stream);
  // hoisted input-gates: xg = ns @ w_ih^T + (b_ih + b_hh), [32000,256]
  biasadd_kernel<<<1, 256, 0, stream>>>(b_ih, b_hh, bsum);
  launch_gemm(ns, 64, w_ih, 64, bsum, xg, 256, 32000, 256, 64, 0, 0, stream);
  lstm_kernel<<<32, 256, 0, stream>>>(xg, w_hh, lstm);
  launch_gemm(lstm, 64, w_in, 64, b_in, qkv, 192, 32000, 192, 64, 0, 0,
              stream);

  const int smemAttn =
      (1008 * KS + 16 * VS + 16 * SS + 16 * 1000 + 8 * 256) * (int)sizeof(float);
  attn_kernel<<<32 * 63, 256, smemAttn, stream>>>(qkv, ctx, attn_out);

  launch_gemm(ctx, 64, w_out, 64, b_out, attended, 64, 32000, 64, 64, 0, 0,
              stream);
  ennpre_kernel<<<32, 256, 0, stream>>>(attended, ennpre);
  launch_gemm(ennpre, 64, W_fix, 64, nullptr, enn, 1024, 32, 1024, 64, 0, 0,
              stream);

  // ---- fusion: fused = [enn|bicep] @ W_fuse^T + b_fuse (two acc passes) ----
  launch_gemm(enn, 1024, W_fuse, 2048, b_fuse, fused, 1024, 32, 1024, 1024, 0,
              1, stream);
  launch_gemm(bicep, 1024, W_fuse + 1024, 2048, nullptr, fused, 1024, 32, 1024,
              1024, 1, 1, stream);
  launch_gemm(fused, 1024, W_proj, 1024, b_proj, out, 1024, 32, 1024, 1024, 0,
              1, stream);
}